// CrossModalHGNN_44890998178033
// MI455X (gfx1250) — compile-verified
//
#include <hip/hip_runtime.h>
#include <math.h>

#define BIcnt 4096
#define BTcnt 4096
#define NNODE 8192
#define DIM   512
#define INV_TAU (1.0f/0.07f)
#define EPS_DEG 1e-6f
#define LN_EPS  1e-5f

typedef __attribute__((ext_vector_type(2))) float v2f;
typedef __attribute__((ext_vector_type(8))) float v8f;

#define WMMA_F32(A, B, C) __builtin_amdgcn_wmma_f32_16x16x4_f32(false, (A), false, (B), (short)0, (C), false, false)

// ---------------- row normalization of features ----------------
__global__ __launch_bounds__(128) void hg_normalize(const float* __restrict__ fimg,
                                                    const float* __restrict__ ftxt,
                                                    float* __restrict__ Fi,
                                                    float* __restrict__ Ft) {
  int r = blockIdx.x, t = threadIdx.x;
  const float* src; float* dst;
  if (r < BIcnt) { src = fimg + (size_t)r * DIM;            dst = Fi + (size_t)r * DIM; }
  else           { src = ftxt + (size_t)(r - BIcnt) * DIM;  dst = Ft + (size_t)(r - BIcnt) * DIM; }
  float x[4]; float ss = 0.f;
#pragma unroll
  for (int u = 0; u < 4; u++) { x[u] = src[t + u * 128]; ss += x[u] * x[u]; }
  __shared__ float red[128];
  red[t] = ss; __syncthreads();
#pragma unroll
  for (int s = 64; s > 0; s >>= 1) { if (t < s) red[t] += red[t + s]; __syncthreads(); }
  float inv = 1.0f / sqrtf(red[0]);
#pragma unroll
  for (int u = 0; u < 4; u++) dst[t + u * 128] = x[u] * inv;
}

// ---------------- S = (Fi @ Ft^T)/TAU, fp32 WMMA, 16x64 strip per wave; writes S and S^T ------
__global__ __launch_bounds__(128) void hg_gemm_s(const float* __restrict__ Fi,
                                                 const float* __restrict__ Ft,
                                                 float* __restrict__ S,
                                                 float* __restrict__ ST) {
  __shared__ float tile[4][16][68];   // per-wave 16x(4*17) padded transpose staging
  int lane = threadIdx.x;             // 0..31 (wave32)
  int wid  = threadIdx.y;             // 0..3
  int tn = (blockIdx.x * 4 + wid) * 64;
  int tm = blockIdx.y * 16;
  int lo  = lane & 15;
  int hi  = lane >> 4;
  int khi = hi * 2;
  const float* ar = Fi + (size_t)(tm + lo) * DIM + khi;
  const float* b0 = Ft + (size_t)(tn + lo) * DIM + khi;
  const float* b1 = b0 + 16 * DIM;
  const float* b2 = b0 + 32 * DIM;
  const float* b3 = b0 + 48 * DIM;
  v8f c0 = {}, c1 = {}, c2 = {}, c3 = {};
#pragma unroll 2
  for (int k = 0; k < DIM; k += 4) {
    v2f a, b;
    a.x = ar[k]; a.y = ar[k + 1];
    b.x = b0[k]; b.y = b0[k + 1];
    c0 = WMMA_F32(a, b, c0);
    b.x = b1[k]; b.y = b1[k + 1];
    c1 = WMMA_F32(a, b, c1);
    b.x = b2[k]; b.y = b2[k + 1];
    c2 = WMMA_F32(a, b, c2);
    b.x = b3[k]; b.y = b3[k + 1];
    c3 = WMMA_F32(a, b, c3);
  }
#pragma unroll
  for (int r = 0; r < 8; r++) {
    int m = r + 8 * hi;
    float v0 = c0[r] * INV_TAU;
    float v1 = c1[r] * INV_TAU;
    float v2 = c2[r] * INV_TAU;
    float v3 = c3[r] * INV_TAU;
    size_t ro = (size_t)(tm + m) * BTcnt + tn + lo;
    S[ro] = v0; S[ro + 16] = v1; S[ro + 32] = v2; S[ro + 48] = v3;
    tile[wid][m][lo] = v0; tile[wid][m][17 + lo] = v1;
    tile[wid][m][34 + lo] = v2; tile[wid][m][51 + lo] = v3;
  }
  __syncthreads();
#pragma unroll
  for (int q = 0; q < 4; q++) {
#pragma unroll
    for (int r = 0; r < 8; r++) {
      int n2 = r + 8 * hi;
      ST[(size_t)(tn + q * 16 + n2) * BIcnt + tm + lo] = tile[wid][lo][q * 17 + n2];
    }
  }
}

// ---------------- top-8 per row (sorted desc) ----------------
__global__ __launch_bounds__(256) void hg_topk(const float* __restrict__ M, int* __restrict__ idxOut) {
  __shared__ float lv[256 * 8];
  __shared__ int   li[256 * 8];
  int r = blockIdx.x, t = threadIdx.x;
  const float* row = M + (size_t)r * BTcnt;
  float v[8]; int id[8];
#pragma unroll
  for (int q = 0; q < 8; q++) { v[q] = -INFINITY; id[q] = 0; }
  for (int j = t; j < BTcnt; j += 256) {
    float x = row[j];
    if (x > v[7]) {
      v[7] = x; id[7] = j;
#pragma unroll
      for (int p = 7; p > 0; --p)
        if (v[p] > v[p - 1]) {
          float tv = v[p]; v[p] = v[p - 1]; v[p - 1] = tv;
          int ti = id[p]; id[p] = id[p - 1]; id[p - 1] = ti;
        }
    }
  }
  for (int stride = 128; stride >= 1; stride >>= 1) {
#pragma unroll
    for (int q = 0; q < 8; q++) { lv[t * 8 + q] = v[q]; li[t * 8 + q] = id[q]; }
    __syncthreads();
    if (t < stride) {
      int p8 = (t + stride) * 8;
#pragma unroll
      for (int q = 0; q < 8; q++) {
        float x = lv[p8 + q]; int xi = li[p8 + q];
        if (x > v[7]) {
          v[7] = x; id[7] = xi;
#pragma unroll
          for (int p = 7; p > 0; --p)
            if (v[p] > v[p - 1]) {
              float tv = v[p]; v[p] = v[p - 1]; v[p - 1] = tv;
              int ti = id[p]; id[p] = id[p - 1]; id[p - 1] = ti;
            }
        }
      }
    }
    __syncthreads();
  }
  if (t == 0) {
#pragma unroll
    for (int q = 0; q < 8; q++) idxOut[(size_t)r * 8 + q] = id[q];
  }
}

// ---------------- small init kernels ----------------
__global__ void hg_dv_init(float* dv) { dv[blockIdx.x * 256 + threadIdx.x] = 2.0f; }
__global__ void hg_dvis(float* dv)    { int g = blockIdx.x * 256 + threadIdx.x; dv[g] = rsqrtf(dv[g] + EPS_DEG); }
__global__ void hg_zero(float* p)     { p[(size_t)blockIdx.x * 256 + threadIdx.x] = 0.0f; }

// ---------------- mutual masking + softmax weights (image-edge columns e = i) ----------------
__global__ __launch_bounds__(256) void hg_edges_img(const float* __restrict__ S,
                                                    const int* __restrict__ idxA,
                                                    const int* __restrict__ idxB,
                                                    int* __restrict__ erow, float* __restrict__ ew,
                                                    float* __restrict__ dv, float* __restrict__ de_inv) {
  int i = blockIdx.x * 256 + threadIdx.x;
  if (i >= BIcnt) return;
  int fin[8]; int m = 0;
#pragma unroll
  for (int k = 0; k < 8; k++) {
    int j = idxA[i * 8 + k];
    bool mut = false;
#pragma unroll
    for (int q = 0; q < 8; q++) mut = mut || (idxB[j * 8 + q] == i);
    if (mut) { fin[m] = j; m++; }
  }
  int j2 = 0;                                  // pad with ascending unused indices (jax NEG tie-break)
  while (m < 8 && j2 < BTcnt) {
    bool used = false;
    for (int q = 0; q < m; q++) used = used || (fin[q] == j2);
    if (!used) { fin[m] = j2; m++; }
    j2++;
  }
  float vals[8]; float mx = -INFINITY;
#pragma unroll
  for (int k = 0; k < 8; k++) { vals[k] = S[(size_t)i * BTcnt + fin[k]]; mx = fmaxf(mx, vals[k]); }
  float sum = 0.f;
#pragma unroll
  for (int k = 0; k < 8; k++) { vals[k] = expf(vals[k] - mx); sum += vals[k]; }
  float inv = 1.0f / sum; float wsum = 0.f;
#pragma unroll
  for (int k = 0; k < 8; k++) {
    float w = vals[k] * inv;
    erow[(size_t)i * 8 + k] = BIcnt + fin[k];
    ew[(size_t)i * 8 + k] = w;
    atomicAdd(&dv[BIcnt + fin[k]], w);
    wsum += w;
  }
  de_inv[i] = 1.0f / (2.0f + wsum + EPS_DEG);
}

// ---------------- text-edge columns e = Bi + j ----------------
__global__ __launch_bounds__(256) void hg_edges_txt(const float* __restrict__ S,
                                                    const int* __restrict__ idxA,
                                                    const int* __restrict__ idxB,
                                                    int* __restrict__ erow, float* __restrict__ ew,
                                                    float* __restrict__ dv, float* __restrict__ de_inv) {
  int j = blockIdx.x * 256 + threadIdx.x;
  if (j >= BTcnt) return;
  int fin[8]; int m = 0;
#pragma unroll
  for (int k = 0; k < 8; k++) {
    int i = idxB[j * 8 + k];
    bool mut = false;
#pragma unroll
    for (int q = 0; q < 8; q++) mut = mut || (idxA[i * 8 + q] == j);
    if (mut) { fin[m] = i; m++; }
  }
  int i2 = 0;
  while (m < 8 && i2 < BIcnt) {
    bool used = false;
    for (int q = 0; q < m; q++) used = used || (fin[q] == i2);
    if (!used) { fin[m] = i2; m++; }
    i2++;
  }
  float vals[8]; float mx = -INFINITY;
#pragma unroll
  for (int k = 0; k < 8; k++) { vals[k] = S[(size_t)fin[k] * BTcnt + j]; mx = fmaxf(mx, vals[k]); }
  float sum = 0.f;
#pragma unroll
  for (int k = 0; k < 8; k++) { vals[k] = expf(vals[k] - mx); sum += vals[k]; }
  float inv = 1.0f / sum; float wsum = 0.f;
  int e = BIcnt + j;
#pragma unroll
  for (int k = 0; k < 8; k++) {
    float w = vals[k] * inv;
    erow[(size_t)e * 8 + k] = fin[k];
    ew[(size_t)e * 8 + k] = w;
    atomicAdd(&dv[fin[k]], w);
    wsum += w;
  }
  de_inv[e] = 1.0f / (2.0f + wsum + EPS_DEG);
}

// ---------------- X init ----------------
__global__ void hg_copy_x(const float* __restrict__ fimg, const float* __restrict__ ftxt, float* __restrict__ X) {
  int v = blockIdx.x, t = threadIdx.x;
  const float* src = (v < BIcnt) ? (fimg + (size_t)v * DIM) : (ftxt + (size_t)(v - BIcnt) * DIM);
  float* dst = X + (size_t)v * DIM;
  dst[t] = src[t]; dst[t + 256] = src[t + 256];
}

// ---------------- Xe = de_inv * (H^T @ (X * dv_is)) : 9-way gather per edge ----------------
__global__ __launch_bounds__(256) void hg_xe(const float* __restrict__ X, const int* __restrict__ erow,
                                             const float* __restrict__ ew, const float* __restrict__ dvis,
                                             const float* __restrict__ de_inv, float* __restrict__ Xe) {
  __shared__ int rs[8]; __shared__ float wk[8];
  int e = blockIdx.x, t = threadIdx.x;
  if (t < 8) { int r = erow[(size_t)e * 8 + t]; rs[t] = r; wk[t] = ew[(size_t)e * 8 + t] * dvis[r]; }
  __syncthreads();
  float sscale = 2.0f * dvis[e];
  float dei = de_inv[e];
#pragma unroll
  for (int u = 0; u < 2; u++) {
    int d = t + u * 256;
    float acc = sscale * X[(size_t)e * DIM + d];
#pragma unroll
    for (int k = 0; k < 8; k++) acc += wk[k] * X[(size_t)rs[k] * DIM + d];
    Xe[(size_t)e * DIM + d] = dei * acc;
  }
}

// ---------------- ACC += scatter of H @ Xe (off-diagonal part) ----------------
__global__ __launch_bounds__(256) void hg_scatter(const float* __restrict__ Xe, const int* __restrict__ erow,
                                                  const float* __restrict__ ew, float* __restrict__ ACC) {
  __shared__ int rs[8]; __shared__ float wk[8];
  int e = blockIdx.x, t = threadIdx.x;
  if (t < 8) { rs[t] = erow[(size_t)e * 8 + t]; wk[t] = ew[(size_t)e * 8 + t]; }
  __syncthreads();
#pragma unroll
  for (int u = 0; u < 2; u++) {
    int d = t + u * 256;
    float xv = Xe[(size_t)e * DIM + d];
#pragma unroll
    for (int k = 0; k < 8; k++) atomicAdd(&ACC[(size_t)rs[k] * DIM + d], wk[k] * xv);
  }
}

// ---------------- X1 = dv_is * (2*Xe + ACC), in place into ACC ----------------
__global__ void hg_finalize(const float* __restrict__ Xe, const float* __restrict__ dvis, float* __restrict__ ACC) {
  size_t g = (size_t)blockIdx.x * 256 + threadIdx.x;
  int v = (int)(g >> 9);
  ACC[g] = dvis[v] * (2.0f * Xe[g] + ACC[g]);
}

// ---------------- Y = gelu(X1 @ W^T), fp32 WMMA, 16x64 strip per wave, exact-GELU epilogue ----
__global__ __launch_bounds__(128) void hg_gemm_fc(const float* __restrict__ X1,
                                                  const float* __restrict__ W,
                                                  float* __restrict__ Y) {
  int lane = threadIdx.x, wid = threadIdx.y;
  int tn = (blockIdx.x * 4 + wid) * 64;
  int tm = blockIdx.y * 16;
  int lo = lane & 15, hi = lane >> 4;
  int khi = hi * 2;
  const float* ar = X1 + (size_t)(tm + lo) * DIM + khi;
  const float* b0 = W  + (size_t)(tn + lo) * DIM + khi;   // B[k][o] = W[o][k]
  const float* b1 = b0 + 16 * DIM;
  const float* b2 = b0 + 32 * DIM;
  const float* b3 = b0 + 48 * DIM;
  v8f c0 = {}, c1 = {}, c2 = {}, c3 = {};
#pragma unroll 2
  for (int k = 0; k < DIM; k += 4) {
    v2f a, b;
    a.x = ar[k]; a.y = ar[k + 1];
    b.x = b0[k]; b.y = b0[k + 1];
    c0 = WMMA_F32(a, b, c0);
    b.x = b1[k]; b.y = b1[k + 1];
    c1 = WMMA_F32(a, b, c1);
    b.x = b2[k]; b.y = b2[k + 1];
    c2 = WMMA_F32(a, b, c2);
    b.x = b3[k]; b.y = b3[k + 1];
    c3 = WMMA_F32(a, b, c3);
  }
#pragma unroll
  for (int r = 0; r < 8; r++) {
    int m = r + 8 * hi;
    size_t ro = (size_t)(tm + m) * DIM + tn + lo;
    float x0 = c0[r], x1 = c1[r], x2 = c2[r], x3 = c3[r];
    Y[ro]      = 0.5f * x0 * (1.0f + erff(x0 * 0.70710678118654752f));
    Y[ro + 16] = 0.5f * x1 * (1.0f + erff(x1 * 0.70710678118654752f));
    Y[ro + 32] = 0.5f * x2 * (1.0f + erff(x2 * 0.70710678118654752f));
    Y[ro + 48] = 0.5f * x3 * (1.0f + erff(x3 * 0.70710678118654752f));
  }
}

// ---------------- X = layernorm(X + Y, g, b) ----------------
__global__ __launch_bounds__(128) void hg_ln_res(float* __restrict__ X, const float* __restrict__ Y,
                                                 const float* __restrict__ g, const float* __restrict__ b) {
  int v = blockIdx.x, t = threadIdx.x;
  float x[4];
#pragma unroll
  for (int u = 0; u < 4; u++) x[u] = X[(size_t)v * DIM + t + u * 128] + Y[(size_t)v * DIM + t + u * 128];
  __shared__ float red[128];
  red[t] = x[0] + x[1] + x[2] + x[3]; __syncthreads();
#pragma unroll
  for (int s = 64; s > 0; s >>= 1) { if (t < s) red[t] += red[t + s]; __syncthreads(); }
  float mu = red[0] * (1.0f / DIM);
  __syncthreads();
  float vs = 0.f;
#pragma unroll
  for (int u = 0; u < 4; u++) { float d = x[u] - mu; vs += d * d; }
  red[t] = vs; __syncthreads();
#pragma unroll
  for (int s = 64; s > 0; s >>= 1) { if (t < s) red[t] += red[t + s]; __syncthreads(); }
  float rs = rsqrtf(red[0] * (1.0f / DIM) + LN_EPS);
#pragma unroll
  for (int u = 0; u < 4; u++) {
    int d = t + u * 128;
    X[(size_t)v * DIM + d] = (x[u] - mu) * rs * g[d] + b[d];
  }
}

// ---------------- final LN into d_out (img params for v<Bi, txt otherwise) ----------------
__global__ __launch_bounds__(128) void hg_ln_out(const float* __restrict__ X,
                                                 const float* __restrict__ gi, const float* __restrict__ bi,
                                                 const float* __restrict__ gt, const float* __restrict__ bt,
                                                 float* __restrict__ out) {
  int v = blockIdx.x, t = threadIdx.x;
  const float* g = (v < BIcnt) ? gi : gt;
  const float* b = (v < BIcnt) ? bi : bt;
  float x[4];
#pragma unroll
  for (int u = 0; u < 4; u++) x[u] = X[(size_t)v * DIM + t + u * 128];
  __shared__ float red[128];
  red[t] = x[0] + x[1] + x[2] + x[3]; __syncthreads();
#pragma unroll
  for (int s = 64; s > 0; s >>= 1) { if (t < s) red[t] += red[t + s]; __syncthreads(); }
  float mu = red[0] * (1.0f / DIM);
  __syncthreads();
  float vs = 0.f;
#pragma unroll
  for (int u = 0; u < 4; u++) { float d = x[u] - mu; vs += d * d; }
  red[t] = vs; __syncthreads();
#pragma unroll
  for (int s = 64; s > 0; s >>= 1) { if (t < s) red[t] += red[t + s]; __syncthreads(); }
  float rs = rsqrtf(red[0] * (1.0f / DIM) + LN_EPS);
#pragma unroll
  for (int u = 0; u < 4; u++) {
    int d = t + u * 128;
    out[(size_t)v * DIM + d] = (x[u] - mu) * rs * g[d] + b[d];
  }
}

extern "C" void kernel_launch(void* const* d_in, const int* in_sizes, int n_in,
                              void* d_out, int out_size, void* d_ws, size_t ws_size,
                              hipStream_t stream) {
  (void)in_sizes; (void)n_in; (void)out_size; (void)ws_size;
  const float* f_img = (const float*)d_in[0];
  const float* f_txt = (const float*)d_in[1];
  const float* Wmat  = (const float*)d_in[2];   // (2, 512, 512)
  const float* ln_g  = (const float*)d_in[3];   // (2, 512)
  const float* ln_b  = (const float*)d_in[4];
  const float* ln_ig = (const float*)d_in[5];
  const float* ln_ib = (const float*)d_in[6];
  const float* ln_tg = (const float*)d_in[7];
  const float* ln_tb = (const float*)d_in[8];

  char* ws = (char*)d_ws;
  size_t o = 0;
  auto alloc = [&](size_t bytes) -> void* {
    void* p = ws + o; o += (bytes + 255) & ~(size_t)255; return p;
  };
  float* Fi = (float*)alloc((size_t)BIcnt * DIM * 4);
  float* Ft = (float*)alloc((size_t)BTcnt * DIM * 4);
  float* S  = (float*)alloc((size_t)BIcnt * BTcnt * 4);
  float* ST = (float*)alloc((size_t)BIcnt * BTcnt * 4);   // reused as X/Xe/ACC/Y after selection
  int*   idxA = (int*)alloc((size_t)BIcnt * 8 * 4);
  int*   idxB = (int*)alloc((size_t)BTcnt * 8 * 4);
  int*   erow = (int*)alloc((size_t)NNODE * 8 * 4);
  float* ew   = (float*)alloc((size_t)NNODE * 8 * 4);
  float* dv   = (float*)alloc((size_t)NNODE * 4);        // becomes dv_is in place
  float* dei  = (float*)alloc((size_t)NNODE * 4);

  // overlay layer buffers on ST (S^T dead after column top-k; 4 x 16 MB == 64 MB == |ST|)
  float* X   = ST;
  float* Xe  = ST + 1 * (size_t)NNODE * DIM;
  float* ACC = ST + 2 * (size_t)NNODE * DIM;
  float* Y   = ST + 3 * (size_t)NNODE * DIM;

  dim3 wb(32, 4);

  hg_normalize<<<NNODE, 128, 0, stream>>>(f_img, f_txt, Fi, Ft);
  hg_gemm_s<<<dim3(BTcnt / 256, BIcnt / 16), wb, 0, stream>>>(Fi, Ft, S, ST);
  hg_topk<<<BIcnt, 256, 0, stream>>>(S, idxA);
  hg_topk<<<BTcnt, 256, 0, stream>>>(ST, idxB);
  hg_dv_init<<<NNODE / 256, 256, 0, stream>>>(dv);
  hg_edges_img<<<BIcnt / 256, 256, 0, stream>>>(S, idxA, idxB, erow, ew, dv, dei);
  hg_edges_txt<<<BTcnt / 256, 256, 0, stream>>>(S, idxA, idxB, erow, ew, dv, dei);
  hg_dvis<<<NNODE / 256, 256, 0, stream>>>(dv);
  hg_copy_x<<<NNODE, 256, 0, stream>>>(f_img, f_txt, X);

  for (int l = 0; l < 2; l++) {
    hg_xe<<<NNODE, 256, 0, stream>>>(X, erow, ew, dv, dei, Xe);
    hg_zero<<<(NNODE * DIM) / 256, 256, 0, stream>>>(ACC);
    hg_scatter<<<NNODE, 256, 0, stream>>>(Xe, erow, ew, ACC);
    hg_finalize<<<(NNODE * DIM) / 256, 256, 0, stream>>>(Xe, dv, ACC);
    hg_gemm_fc<<<dim3(DIM / 256, NNODE / 16), wb, 0, stream>>>(ACC, Wmat + (size_t)l * DIM * DIM, Y);
    hg_ln_res<<<NNODE, 128, 0, stream>>>(X, Y, ln_g + (size_t)l * DIM, ln_b + (size_t)l * DIM);
  }

  hg_ln_out<<<NNODE, 128, 0, stream>>>(X, ln_ig, ln_ib, ln_tg, ln_tb, (float*)d_out);
}